// HyperbolicCrossEntropyLoss_32590211842285
// MI455X (gfx1250) — compile-verified
//
#include <hip/hip_runtime.h>
#include <hip/hip_bf16.h>

// Hyperbolic cross-entropy loss for MI455X (gfx1250, wave32).
//
// Strategy:
//   dist(x,y) depends only on |x|^2, |y|^2, <x,y>  (Mobius-add norm collapses
//   to scalars), and expmap0+project is a uniform per-row scale. So the heavy
//   part is a [N=262144, C=64] x [C=64, K=19] fp32 GEMM which we run on
//   V_WMMA_F32_16X16X4_F32 (16 pixels/wave, two 16-wide class tiles, 16
//   K-steps). Everything else (expmap scale, atanh distance, log-softmax over
//   19 classes, masked NLL mean) is a cheap VALU/shuffle epilogue.
//
// Roofline: 67 MB streamed @ 23.3 TB/s ~ 2.9 us; ~0.64 GFLOP of MACs through
// fp32 WMMA keeps compute below the memory ceiling (pure VALU fp32 would be
// borderline). fp32 WMMA also matches the fp32 inputs exactly (no cvt cost).

#define NUM_K     19
#define CHANNELS  64
#define HW        65536      // 256*256
#define NPIX      262144     // 4*256*256
#define PSTRIDE   66         // LDS row stride (even for b64 loads, conflict-poor)
#define EPSF      1e-15f
#define MAXNORM   (1.0f - 1e-5f)
#define IGN       255

typedef float v2f __attribute__((ext_vector_type(2)));
typedef float v8f __attribute__((ext_vector_type(8)));

// expmap0 + _project collapse to a uniform scale s; returns s, writes |x_hyp|^2.
__device__ __forceinline__ float expmap_scale(float n2, float* x2h) {
    float n  = sqrtf(n2);
    float t  = tanhf(n);
    float s  = t / (n + EPSF);
    float vn = s * n;                       // = |expmap0(x)|
    if (vn > MAXNORM) s *= MAXNORM / (vn + EPSF);
    vn = s * n;
    *x2h = vn * vn;
    return s;
}

// logit = -dist for one (pixel, class) pair from the three scalars.
__device__ __forceinline__ float hyp_logit(float rawdot, float sx, float x2, float y2) {
    float xy   = -(sx * rawdot);            // <(-x_hyp), y_hyp>
    float A    = 1.0f + 2.0f * xy + y2;     // c = 1
    float Bq   = 1.0f - x2;
    float num2 = A * A * x2 + 2.0f * A * Bq * xy + Bq * Bq * y2;
    float den  = 1.0f + 2.0f * xy + x2 * y2;
    float d    = sqrtf(fmaxf(num2, 0.0f)) / (den + EPSF);
    d = fminf(fmaxf(d, -1.0f + 1e-7f), 1.0f - 1e-7f);
    // dist = 2*atanh(d) = log((1+d)/(1-d))
    return -__logf((1.0f + d) / (1.0f - d));
}

__global__ void hyploss_init(float* acc) {
    if (threadIdx.x < 2) acc[threadIdx.x] = 0.0f;
}

__global__ __launch_bounds__(256) void hyploss_main(
        const float* __restrict__ x,        // [4,64,256,256] NCHW
        const int*   __restrict__ target,   // [4,256,256]
        const float* __restrict__ protos,   // [19,64]
        float*       __restrict__ acc) {    // acc[0]=sum nll, acc[1]=count

    __shared__ float sP[32 * PSTRIDE];      // expmapped prototypes, rows 19..31 zero
    __shared__ float sY2[32];               // |proto_hyp|^2, zero-padded
    __shared__ float sSum[8], sCnt[8];

    const int tid = threadIdx.x;

    // ---- prologue: build proto_hyp in LDS (tiny: 19 x 64) ----
    for (int i = tid; i < 32 * PSTRIDE; i += 256) sP[i] = 0.0f;
    if (tid < 32) sY2[tid] = 0.0f;
    __syncthreads();
    if (tid < NUM_K) {
        float n2 = 0.0f;
        for (int c = 0; c < CHANNELS; ++c) {
            float v = protos[tid * CHANNELS + c];
            n2 += v * v;
        }
        float y2; float s = expmap_scale(n2, &y2);
        for (int c = 0; c < CHANNELS; ++c)
            sP[tid * PSTRIDE + c] = s * protos[tid * CHANNELS + c];
        sY2[tid] = y2;
    }
    __syncthreads();

    const int lane = tid & 31;
    const int wave = tid >> 5;
    const int half = lane >> 4;             // 0: K-cols {0,1} of 4; 1: {2,3}
    const int l16  = lane & 15;

    const int p0    = blockIdx.x * 128 + wave * 16;  // this wave's 16 pixels
    const int myPix = p0 + l16;                      // pixel owned by lane (A rows)
    const int b     = myPix >> 16;                   // myPix / HW
    const int hw    = myPix & (HW - 1);
    const int base  = b * (CHANNELS * HW) + hw;      // fits in int (<2^25)

    // ---- GEMM: D[16x16] += A[16x4] * B[4x16], 16 K-steps, 2 class tiles ----
    v8f c0 = {};    // classes 0..15
    v8f c1 = {};    // classes 16..31 (19..31 are zero columns)
    float rawx2 = 0.0f;

    #pragma unroll
    for (int s = 0; s < 16; ++s) {
        const int ch = 4 * s + 2 * half;
        v2f a;
        a.x = x[base + ch * HW];            // coalesced: lanes share channel, walk w
        a.y = x[base + (ch + 1) * HW];
        rawx2 += a.x * a.x + a.y * a.y;     // per-lane half of |x|^2
        const v2f b0 = *(const v2f*)&sP[l16 * PSTRIDE + ch];
        const v2f b1 = *(const v2f*)&sP[(16 + l16) * PSTRIDE + ch];
        c0 = __builtin_amdgcn_wmma_f32_16x16x4_f32(false, a, false, b0,
                                                   (short)0, c0, false, false);
        c1 = __builtin_amdgcn_wmma_f32_16x16x4_f32(false, a, false, b1,
                                                   (short)0, c1, false, false);
    }

    // combine channel halves: lane now holds |x|^2 of pixel p0+l16
    rawx2 += __shfl_xor(rawx2, 16, 32);
    float x2h; float sx = expmap_scale(rawx2, &x2h);
    int   tgt = target[myPix];

    const float y2_0   = sY2[l16];
    const float y2_1   = sY2[16 + l16];
    const bool  valid1 = (l16 < (NUM_K - 16));       // classes 16..18

    float accSum = 0.0f, accCnt = 0.0f;

    // ---- epilogue: C layout is (pixel = r + 8*half, class = l16 [+16]) ----
    #pragma unroll
    for (int r = 0; r < 8; ++r) {
        const int srcLane = r + 8 * half;            // lane owning this pixel's scalars
        const float x2p  = __shfl(x2h, srcLane, 32);
        const float sxp  = __shfl(sx,  srcLane, 32);
        const int   tgtp = __shfl(tgt, srcLane, 32);

        const float l0 = hyp_logit(c0[r], sxp, x2p, y2_0);
        const float l1 = hyp_logit(c1[r], sxp, x2p, y2_1);
        const float l1m = valid1 ? l1 : -3.0e38f;

        // log-softmax over 19 classes: reduce across the 16-lane half
        float m = fmaxf(l0, l1m);
        #pragma unroll
        for (int o = 1; o < 16; o <<= 1) m = fmaxf(m, __shfl_xor(m, o, 32));

        float e  = __expf(l0 - m) + (valid1 ? __expf(l1 - m) : 0.0f);
        float lt = ((l16 == tgtp) ? l0 : 0.0f)
                 + ((valid1 && (16 + l16 == tgtp)) ? l1 : 0.0f);
        #pragma unroll
        for (int o = 1; o < 16; o <<= 1) {
            e  += __shfl_xor(e,  o, 32);
            lt += __shfl_xor(lt, o, 32);
        }
        const float lse = m + __logf(e);

        if (l16 == 0) {                              // one lane per half accumulates
            const bool valid = (tgtp != IGN);
            accSum += valid ? (lse - lt) : 0.0f;
            accCnt += valid ? 1.0f : 0.0f;
        }
    }

    // ---- block reduction + global accumulate ----
    accSum += __shfl_xor(accSum, 16, 32);
    accCnt += __shfl_xor(accCnt, 16, 32);
    if (lane == 0) { sSum[wave] = accSum; sCnt[wave] = accCnt; }
    __syncthreads();
    if (tid == 0) {
        float S = 0.0f, Cn = 0.0f;
        #pragma unroll
        for (int w = 0; w < 8; ++w) { S += sSum[w]; Cn += sCnt[w]; }
        atomicAdd(&acc[0], S);
        atomicAdd(&acc[1], Cn);
    }
}

__global__ void hyploss_finalize(const float* __restrict__ acc,
                                 float* __restrict__ out) {
    out[0] = acc[0] / fmaxf(acc[1], 1.0f);
}

extern "C" void kernel_launch(void* const* d_in, const int* in_sizes, int n_in,
                              void* d_out, int out_size, void* d_ws, size_t ws_size,
                              hipStream_t stream) {
    const float* x       = (const float*)d_in[0];
    const int*   target  = (const int*)d_in[1];
    const float* protos  = (const float*)d_in[2];
    float*       acc     = (float*)d_ws;           // 2 floats: sum, count
    float*       out     = (float*)d_out;

    hyploss_init<<<1, 32, 0, stream>>>(acc);
    hyploss_main<<<NPIX / 128, 256, 0, stream>>>(x, target, protos, acc);
    hyploss_finalize<<<1, 1, 0, stream>>>(acc, out);
}